// TransformerConv_68049461837966
// MI455X (gfx1250) — compile-verified
//
#include <hip/hip_runtime.h>
#include <math.h>
#include <stdint.h>

// ---------------------------------------------------------------------------
// TransformerConv (PyG-style) for MI455X / gfx1250 (wave32, WMMA).
//
// Sizes: N=50000 nodes, E=800000 edges, NIN=128, H=8 heads, C=16 -> H*C=128.
//
// Memory-bound workload (~0.9 GB HBM traffic, ~33 GFLOP fp32). Strategy:
//  - All GEMMs via V_WMMA_F32_16X16X4_F32 (full fp32 path, 16x16 tiles).
//  - Tile staging uses GLOBAL_LOAD_ASYNC_TO_LDS_B128 (CDNA5 async path,
//    ASYNCcnt) -- no VGPR landing, half the instructions of load+ds_store.
//  - e = edge_attr @ We is RECOMPUTED in the scatter pass instead of being
//    materialized (saves 409.6 MB of HBM write+read and 409.6 MB workspace).
//  - alpha is fused into the edge GEMM: wave w's 16-col tile == head w's
//    channels, so the per-head dot product reduces inside one wave.
// ---------------------------------------------------------------------------

#define NNODES 50000
#define NEDGES 800000
#define KDIM   128   // NIN
#define HC     128   // H*C
#define NH     8
#define XS_STRIDE 132  // LDS row stride in floats; 128 would 64-bank conflict
                       // (132*4 = 528 B = 33*16 B, so float4 slots stay 16B-aligned)

typedef __attribute__((ext_vector_type(2))) float v2f;
typedef __attribute__((ext_vector_type(8))) float v8f;

// ---- stage a 16x128 fp32 tile into LDS via async global->LDS DMA ----------
// Each of the 256 threads moves two 16B chunks straight into LDS
// (global_load_async_to_lds_b128, tracked by ASYNCcnt; no VGPR landing).
__device__ __forceinline__ void stage_tile(const float* __restrict__ src,
                                           float* __restrict__ xs, int t) {
#pragma unroll
    for (int i = 0; i < 2; ++i) {
        int idx = t + i * 256;          // 0..511 float4 slots
        int r   = idx >> 5;             // row 0..15
        int c4  = idx & 31;             // float4 col 0..31
        const float* g = src + r * KDIM + c4 * 4;
        // low 32 bits of the flat shared address == LDS byte address
        uint32_t lds = (uint32_t)(uintptr_t)&xs[r * XS_STRIDE + c4 * 4];
        asm volatile("global_load_async_to_lds_b128 %0, %1, off"
                     :
                     : "v"(lds), "v"(g)
                     : "memory");
    }
    asm volatile("s_wait_asynccnt 0x0" ::: "memory");  // this wave's DMAs done
}

// ---- 16x16 fp32 WMMA GEMM body: C += xs[16x128] * W[:,n0:n0+16] -----------
// A layout (16x4 f32): lane<16 -> M=lane, K=k+{0,1}; lane>=16 -> K=k+{2,3}
// B layout (4x16 f32): lane<16 -> N=lane, K rows symmetric to A.
__device__ __forceinline__ v8f gemm_tile(const float* __restrict__ xs,
                                         const float* __restrict__ W,
                                         int cl, int h2, int n0) {
    const float* xrow = &xs[cl * XS_STRIDE + 2 * h2];
    const float* wp   = W + (2 * h2) * HC + n0 + cl;
    v8f c = {};
#pragma unroll
    for (int kk = 0; kk < KDIM; kk += 4) {
        v2f a, b;
        a.x = xrow[kk];
        a.y = xrow[kk + 1];
        b.x = wp[kk * HC];
        b.y = wp[kk * HC + HC];
        c = __builtin_amdgcn_wmma_f32_16x16x4_f32(
                /*neg_a=*/false, a, /*neg_b=*/false, b,
                /*c_mod=*/(short)0, c, /*reuse_a=*/false, /*reuse_b=*/false);
    }
    return c;
}

// ---------------------------------------------------------------------------
// Kernel 1: q/k/v/skip projections.  grid = (N/16, 4); block = 256 (8 waves).
// grid.y selects which weight; y==3 writes skip (+bias) straight into d_out.
// ---------------------------------------------------------------------------
__global__ __launch_bounds__(256)
void node_gemm_kernel(const float* __restrict__ X,
                      const float* __restrict__ Wq, const float* __restrict__ bq,
                      const float* __restrict__ Wk, const float* __restrict__ bk,
                      const float* __restrict__ Wv, const float* __restrict__ bv,
                      const float* __restrict__ Ws, const float* __restrict__ bs,
                      float* __restrict__ q, float* __restrict__ k,
                      float* __restrict__ v, float* __restrict__ out) {
    __shared__ float xs[16 * XS_STRIDE];
    const float* W; const float* B; float* O;
    switch (blockIdx.y) {
        case 0:  W = Wq; B = bq; O = q;   break;
        case 1:  W = Wk; B = bk; O = k;   break;
        case 2:  W = Wv; B = bv; O = v;   break;
        default: W = Ws; B = bs; O = out; break;
    }
    const int t  = threadIdx.x;
    const int m0 = blockIdx.x * 16;
    __builtin_prefetch(W, 0, 0);                 // global_prefetch_b8 (L2 warm)
    stage_tile(X + (size_t)m0 * KDIM, xs, t);
    __syncthreads();

    const int lane = t & 31, wv = t >> 5;
    const int cl = lane & 15, h2 = lane >> 4, n0 = wv * 16;
    v8f c = gemm_tile(xs, W, cl, h2, n0);

    const float bb = B[n0 + cl];
#pragma unroll
    for (int r = 0; r < 8; ++r) {                // D: lane<16 M=r, else M=r+8
        int row = m0 + r + 8 * h2;
        O[(size_t)row * HC + n0 + cl] = c[r] + bb;
    }
}

// ---------------------------------------------------------------------------
// Kernel 2: e-tile GEMM fused with alpha.  Wave w == head w (cols 16w..16w+15).
// alpha[e][h] = dot(q[dst], k[src]+e) / sqrt(16)
// ---------------------------------------------------------------------------
__global__ __launch_bounds__(256)
void edge_gemm_alpha_kernel(const float* __restrict__ EA,
                            const float* __restrict__ We,
                            const int* __restrict__ ei,
                            const float* __restrict__ q,
                            const float* __restrict__ k,
                            float* __restrict__ alpha) {
    __shared__ float xs[16 * XS_STRIDE];
    __shared__ int s_src[16], s_dst[16];
    const int tE = blockIdx.x * 16;
    const int t  = threadIdx.x;
    if (t < 16) { s_src[t] = ei[tE + t]; s_dst[t] = ei[NEDGES + tE + t]; }
    stage_tile(EA + (size_t)tE * KDIM, xs, t);
    __syncthreads();

    const int lane = t & 31, wv = t >> 5;        // wv == head index
    const int cl = lane & 15, h2 = lane >> 4, n0 = wv * 16;
    v8f c = gemm_tile(xs, We, cl, h2, n0);

#pragma unroll
    for (int r = 0; r < 8; ++r) {
        int idx = r + 8 * h2;                    // edge slot within tile
        int s = s_src[idx], d = s_dst[idx];
        float kv = k[(size_t)s * HC + n0 + cl];  // gathers hit L2 (q,k fit)
        float qv = q[(size_t)d * HC + n0 + cl];
        float p  = qv * (kv + c[r]);
        // reduce over the 16 channels of this head (one 16-lane half-group)
        p += __shfl_xor(p, 1, 32);
        p += __shfl_xor(p, 2, 32);
        p += __shfl_xor(p, 4, 32);
        p += __shfl_xor(p, 8, 32);
        if (cl == 0) alpha[(size_t)(tE + idx) * NH + wv] = p * 0.25f;
    }
}

// ---------------------------------------------------------------------------
// Softmax bookkeeping kernels
// ---------------------------------------------------------------------------
__global__ void init_mz_kernel(float* __restrict__ m, float* __restrict__ z, int n) {
    int i = blockIdx.x * blockDim.x + threadIdx.x;
    if (i < n) { m[i] = -3.402823466e38f; z[i] = 0.0f; }
}

__device__ __forceinline__ void atomic_max_f32(float* addr, float val) {
    // IEEE ordered-bits trick -> native global_atomic_{max_i32,min_u32}
    if (val >= 0.0f) atomicMax((int*)addr, __float_as_int(val));
    else             atomicMin((unsigned int*)addr, (unsigned int)__float_as_int(val));
}

__global__ void seg_max_kernel(const float* __restrict__ alpha,
                               const int* __restrict__ ei,
                               float* __restrict__ m) {
    int i = blockIdx.x * blockDim.x + threadIdx.x;
    if (i >= NEDGES * NH) return;
    int e = i >> 3, h = i & 7;
    int d = ei[NEDGES + e];
    atomic_max_f32(&m[d * NH + h], alpha[i]);
}

__global__ void exp_z_kernel(float* __restrict__ alpha,
                             const int* __restrict__ ei,
                             const float* __restrict__ m,
                             float* __restrict__ z) {
    int i = blockIdx.x * blockDim.x + threadIdx.x;
    if (i >= NEDGES * NH) return;
    int e = i >> 3, h = i & 7;
    int d = ei[NEDGES + e];
    float a = __expf(alpha[i] - m[d * NH + h]);
    alpha[i] = a;                                 // overwrite with exp value
    atomicAdd(&z[d * NH + h], a);
}

// ---------------------------------------------------------------------------
// Kernel 3: recompute e-tile with WMMA, then out[dst] += (v[src]+e) * a_norm.
// ---------------------------------------------------------------------------
__global__ __launch_bounds__(256)
void edge_msg_kernel(const float* __restrict__ EA,
                     const float* __restrict__ We,
                     const int* __restrict__ ei,
                     const float* __restrict__ v,
                     const float* __restrict__ alpha,
                     const float* __restrict__ z,
                     float* __restrict__ out) {
    __shared__ float xs[16 * XS_STRIDE];
    __shared__ int s_src[16], s_dst[16];
    __shared__ float anorm[16][NH];
    const int tE = blockIdx.x * 16;
    const int t  = threadIdx.x;
    if (t < 16) { s_src[t] = ei[tE + t]; s_dst[t] = ei[NEDGES + tE + t]; }
    stage_tile(EA + (size_t)tE * KDIM, xs, t);
    __syncthreads();

    if (t < 16 * NH) {                            // normalize attention in LDS
        int i = t >> 3, h = t & 7;
        int d = s_dst[i];
        anorm[i][h] = alpha[(size_t)(tE + i) * NH + h] / (z[d * NH + h] + 1e-16f);
    }
    __syncthreads();

    const int lane = t & 31, wv = t >> 5;         // wv == head
    const int cl = lane & 15, h2 = lane >> 4, n0 = wv * 16;
    v8f c = gemm_tile(xs, We, cl, h2, n0);        // recomputed e-tile

#pragma unroll
    for (int r = 0; r < 8; ++r) {
        int idx = r + 8 * h2;
        int s = s_src[idx], d = s_dst[idx];
        float vv  = v[(size_t)s * HC + n0 + cl];
        float msg = (vv + c[r]) * anorm[idx][wv]; // LDS broadcast, no conflict
        atomicAdd(&out[(size_t)d * HC + n0 + cl], msg);
    }
}

// ---------------------------------------------------------------------------
// Host-side launch.  Workspace layout (floats):
//   q[N*128] | k[N*128] | v[N*128] | alpha[E*8] | m[N*8] | z[N*8]  ~106 MB
// ---------------------------------------------------------------------------
extern "C" void kernel_launch(void* const* d_in, const int* in_sizes, int n_in,
                              void* d_out, int out_size, void* d_ws, size_t ws_size,
                              hipStream_t stream) {
    (void)in_sizes; (void)n_in; (void)out_size; (void)ws_size;
    const float* x     = (const float*)d_in[0];
    const int*   ei    = (const int*)  d_in[1];
    const float* ea    = (const float*)d_in[2];
    const float* Wq    = (const float*)d_in[3];
    const float* bq    = (const float*)d_in[4];
    const float* Wk    = (const float*)d_in[5];
    const float* bk    = (const float*)d_in[6];
    const float* Wv    = (const float*)d_in[7];
    const float* bv    = (const float*)d_in[8];
    const float* We    = (const float*)d_in[9];
    const float* Wskip = (const float*)d_in[10];
    const float* bskip = (const float*)d_in[11];
    float* out = (float*)d_out;

    float* q     = (float*)d_ws;
    float* k     = q     + (size_t)NNODES * HC;
    float* v     = k     + (size_t)NNODES * HC;
    float* alpha = v     + (size_t)NNODES * HC;
    float* m     = alpha + (size_t)NEDGES * NH;
    float* z     = m     + (size_t)NNODES * NH;

    // 1) q,k,v into ws; skip+bias into d_out (scatter later accumulates on top)
    node_gemm_kernel<<<dim3(NNODES / 16, 4), 256, 0, stream>>>(
        x, Wq, bq, Wk, bk, Wv, bv, Wskip, bskip, q, k, v, out);

    // 2) softmax state init
    init_mz_kernel<<<(NNODES * NH + 255) / 256, 256, 0, stream>>>(m, z, NNODES * NH);

    // 3) edge GEMM fused with per-head attention logits
    edge_gemm_alpha_kernel<<<NEDGES / 16, 256, 0, stream>>>(ea, We, ei, q, k, alpha);

    // 4) segment max, 5) exp + segment sum
    seg_max_kernel<<<(NEDGES * NH + 255) / 256, 256, 0, stream>>>(alpha, ei, m);
    exp_z_kernel <<<(NEDGES * NH + 255) / 256, 256, 0, stream>>>(alpha, ei, m, z);

    // 6) recompute e-tile + weighted scatter into out
    edge_msg_kernel<<<NEDGES / 16, 256, 0, stream>>>(ea, We, ei, v, alpha, z, out);
}